// GraphSage0TPK_48155173323149
// MI455X (gfx1250) — compile-verified
//
#include <hip/hip_runtime.h>
#include <hip/hip_bf16.h>
#include <math.h>

typedef float v2f __attribute__((ext_vector_type(2)));
typedef float v8f __attribute__((ext_vector_type(8)));

#define FDIM 128  // feature width for all 3 SAGE layers (F_IN == H == 128)

// ---------------- counting / inverse ----------------
__global__ void gs_count_kernel(const int* __restrict__ ids, int* __restrict__ cnt, int n) {
    int i = blockIdx.x * blockDim.x + threadIdx.x;
    if (i < n) atomicAdd(&cnt[ids[i]], 1);
}

__global__ void gs_inv_kernel(const int* __restrict__ cnt, float* __restrict__ inv, int n) {
    int i = blockIdx.x * blockDim.x + threadIdx.x;
    if (i < n) inv[i] = 1.0f / fmaxf((float)cnt[i], 1.0f);
}

// ---------------- edge scatter-add: one wave per edge, 32 lanes x float4 ----------------
__global__ __launch_bounds__(256)
void gs_scatter_kernel(const float* __restrict__ x, const int* __restrict__ src,
                       const int* __restrict__ dst, float* __restrict__ aggr, int E) {
    int wave = (blockIdx.x * blockDim.x + threadIdx.x) >> 5;
    int lane = threadIdx.x & 31;
    if (wave >= E) return;
    int s = src[wave];
    int d = dst[wave];
    const float4* xs = (const float4*)(x + (size_t)s * FDIM);
    float4 v = xs[lane];
    float* out = aggr + (size_t)d * FDIM + lane * 4;
    atomicAdd(out + 0, v.x);
    atomicAdd(out + 1, v.y);
    atomicAdd(out + 2, v.z);
    atomicAdd(out + 3, v.w);
}

// ---------------- fused SAGE layer: out = relu( (aggr*inv_deg)@Wl + x@Wr + bl ) ----------------
// One wave32 per 16x16 output tile, K=128 stepped by 4 with V_WMMA_F32_16X16X4_F32.
// A layout (16x4 f32): lanes 0-15 -> M=lane, K={k,k+1}; lanes 16-31 -> M=lane-16, K={k+2,k+3}.
// B layout (4x16 f32): mirrored with N in the lane dimension.
// C/D layout (16x16 f32, 8 VGPRs): VGPR r = rows {r, r+8}, N = lane%16.
__global__ __launch_bounds__(256)
void gs_sage_wmma_kernel(const float* __restrict__ xin, const float* __restrict__ aggr,
                         const float* __restrict__ inv_deg,
                         const float* __restrict__ Wl, const float* __restrict__ bl,
                         const float* __restrict__ Wr,
                         float* __restrict__ out, int n_tiles) {
    int wave = threadIdx.x >> 5;
    int lane = threadIdx.x & 31;
    int tile = blockIdx.x * 8 + wave;
    if (tile >= n_tiles) return;                 // wave-uniform: EXEC all-ones past here

    const int col_tiles = FDIM / 16;             // 8
    int row_tile = tile / col_tiles;
    int col_tile = tile % col_tiles;

    int lm = lane & 15;
    int hi = lane >> 4;                          // selects K pair {0,1} vs {2,3}
    int row = row_tile * 16 + lm;                // A-matrix row for this lane
    int col = col_tile * 16 + lm;                // B/C/D column for this lane

    float id = inv_deg[row];
    const float* arow = aggr + (size_t)row * FDIM;
    const float* xrow = xin  + (size_t)row * FDIM;

    v8f acc = {};
#pragma unroll 4
    for (int k = 0; k < FDIM; k += 4) {
        int ka = k + 2 * hi;
        float2 am = *(const float2*)(arow + ka);
        float2 ax = *(const float2*)(xrow + ka);
        v2f a_m, a_x, b_l, b_r;
        a_m.x = am.x * id;  a_m.y = am.y * id;
        a_x.x = ax.x;       a_x.y = ax.y;
        b_l.x = Wl[(size_t)ka * FDIM + col];
        b_l.y = Wl[(size_t)(ka + 1) * FDIM + col];
        b_r.x = Wr[(size_t)ka * FDIM + col];
        b_r.y = Wr[(size_t)(ka + 1) * FDIM + col];
        acc = __builtin_amdgcn_wmma_f32_16x16x4_f32(false, a_m, false, b_l,
                                                    (short)0, acc, false, false);
        acc = __builtin_amdgcn_wmma_f32_16x16x4_f32(false, a_x, false, b_r,
                                                    (short)0, acc, false, false);
    }

    float bias = bl[col];
#pragma unroll
    for (int r = 0; r < 8; ++r) {
        int orow = row_tile * 16 + r + 8 * hi;   // C/D layout: VGPR r holds rows r and r+8
        float v = acc[r] + bias;
        v = fmaxf(v, 0.0f);
        out[(size_t)orow * FDIM + col] = v;
    }
}

// ---------------- global mean pool accumulation: wave per node ----------------
__global__ __launch_bounds__(256)
void gs_pool_kernel(const float* __restrict__ h, const int* __restrict__ batch,
                    float* __restrict__ gsum, int N) {
    int wave = (blockIdx.x * blockDim.x + threadIdx.x) >> 5;
    int lane = threadIdx.x & 31;
    if (wave >= N) return;
    int b = batch[wave];
    const float4* hr = (const float4*)(h + (size_t)wave * FDIM);
    float4 v = hr[lane];
    float* out = gsum + (size_t)b * FDIM + lane * 4;
    atomicAdd(out + 0, v.x);
    atomicAdd(out + 1, v.y);
    atomicAdd(out + 2, v.z);
    atomicAdd(out + 3, v.w);
}

// ---------------- MLP head + log_softmax: one block (64 threads) per graph ----------------
__global__ __launch_bounds__(64)
void gs_head_kernel(const float* __restrict__ gsum, const float* __restrict__ inv_bat,
                    const float* __restrict__ W4, const float* __restrict__ b4,
                    const float* __restrict__ W5, const float* __restrict__ b5,
                    float* __restrict__ out, int C) {
    __shared__ float sg[FDIM];
    __shared__ float hid[64];
    __shared__ float lg[16];
    __shared__ float red[2];
    int g = blockIdx.x;
    int t = threadIdx.x;

    float ib = inv_bat[g];
    sg[t]      = gsum[(size_t)g * FDIM + t]      * ib;
    sg[t + 64] = gsum[(size_t)g * FDIM + 64 + t] * ib;
    __syncthreads();

    float acc = b4[t];
#pragma unroll 8
    for (int k = 0; k < FDIM; ++k) acc += sg[k] * W4[(size_t)k * 64 + t];
    hid[t] = fmaxf(acc, 0.0f);
    __syncthreads();

    if (t < C) {
        float l = b5[t];
#pragma unroll 8
        for (int k = 0; k < 64; ++k) l += hid[k] * W5[(size_t)k * C + t];
        lg[t] = l;
    }
    __syncthreads();

    if (t == 0) {
        float m = -3.0e38f;
        for (int c = 0; c < C; ++c) m = fmaxf(m, lg[c]);
        float s = 0.0f;
        for (int c = 0; c < C; ++c) s += expf(lg[c] - m);
        red[0] = m;
        red[1] = logf(s);
    }
    __syncthreads();

    if (t < C) out[(size_t)g * C + t] = lg[t] - red[0] - red[1];
}

extern "C" void kernel_launch(void* const* d_in, const int* in_sizes, int n_in,
                              void* d_out, int out_size, void* d_ws, size_t ws_size,
                              hipStream_t stream) {
    const float* x     = (const float*)d_in[0];
    const int*   eidx  = (const int*)d_in[1];
    const int*   batch = (const int*)d_in[2];
    const float* Wl1 = (const float*)d_in[3];  const float* bl1 = (const float*)d_in[4];  const float* Wr1 = (const float*)d_in[5];
    const float* Wl2 = (const float*)d_in[6];  const float* bl2 = (const float*)d_in[7];  const float* Wr2 = (const float*)d_in[8];
    const float* Wl3 = (const float*)d_in[9];  const float* bl3 = (const float*)d_in[10]; const float* Wr3 = (const float*)d_in[11];
    const float* W4  = (const float*)d_in[12]; const float* b4  = (const float*)d_in[13];
    const float* W5  = (const float*)d_in[14]; const float* b5  = (const float*)d_in[15];

    const int N = in_sizes[0] / FDIM;
    const int E = in_sizes[1] / 2;
    const int C = in_sizes[15];        // b5 length = num classes
    const int G = out_size / C;

    const int* src = eidx;
    const int* dst = eidx + E;

    // ---- workspace carve-out (256B aligned) ----
    char* wsp = (char*)d_ws;
    auto carve = [&](size_t bytes) -> char* {
        char* p = wsp;
        wsp += (bytes + 255) & ~(size_t)255;
        return p;
    };
    float* aggr = (float*)carve((size_t)N * FDIM * sizeof(float));
    float* hA   = (float*)carve((size_t)N * FDIM * sizeof(float));
    float* hB   = (float*)carve((size_t)N * FDIM * sizeof(float));
    float* gsum = (float*)carve((size_t)G * FDIM * sizeof(float));
    int*   cntd = (int*)  carve((size_t)N * sizeof(int));
    int*   cntb = (int*)  carve((size_t)G * sizeof(int));
    float* invd = (float*)carve((size_t)N * sizeof(float));
    float* invb = (float*)carve((size_t)G * sizeof(float));

    // ---- degree and batch counts ----
    hipMemsetAsync(cntd, 0, (size_t)N * sizeof(int), stream);
    hipMemsetAsync(cntb, 0, (size_t)G * sizeof(int), stream);
    gs_count_kernel<<<(E + 255) / 256, 256, 0, stream>>>(dst, cntd, E);
    gs_count_kernel<<<(N + 255) / 256, 256, 0, stream>>>(batch, cntb, N);
    gs_inv_kernel<<<(N + 255) / 256, 256, 0, stream>>>(cntd, invd, N);
    gs_inv_kernel<<<(G + 255) / 256, 256, 0, stream>>>(cntb, invb, G);

    const int n_tiles = (N / 16) * (FDIM / 16);
    const int sage_blocks = (n_tiles + 7) / 8;
    const int scat_blocks = (E + 7) / 8;

    // ---- layer 1: x -> hA ----
    hipMemsetAsync(aggr, 0, (size_t)N * FDIM * sizeof(float), stream);
    gs_scatter_kernel<<<scat_blocks, 256, 0, stream>>>(x, src, dst, aggr, E);
    gs_sage_wmma_kernel<<<sage_blocks, 256, 0, stream>>>(x, aggr, invd, Wl1, bl1, Wr1, hA, n_tiles);

    // ---- layer 2: hA -> hB ----
    hipMemsetAsync(aggr, 0, (size_t)N * FDIM * sizeof(float), stream);
    gs_scatter_kernel<<<scat_blocks, 256, 0, stream>>>(hA, src, dst, aggr, E);
    gs_sage_wmma_kernel<<<sage_blocks, 256, 0, stream>>>(hA, aggr, invd, Wl2, bl2, Wr2, hB, n_tiles);

    // ---- layer 3: hB -> hA ----
    hipMemsetAsync(aggr, 0, (size_t)N * FDIM * sizeof(float), stream);
    gs_scatter_kernel<<<scat_blocks, 256, 0, stream>>>(hB, src, dst, aggr, E);
    gs_sage_wmma_kernel<<<sage_blocks, 256, 0, stream>>>(hB, aggr, invd, Wl3, bl3, Wr3, hA, n_tiles);

    // ---- global mean pool + head ----
    hipMemsetAsync(gsum, 0, (size_t)G * FDIM * sizeof(float), stream);
    gs_pool_kernel<<<(N + 7) / 8, 256, 0, stream>>>(hA, batch, gsum, N);
    gs_head_kernel<<<G, 64, 0, stream>>>(gsum, invb, W4, b4, W5, b5, (float*)d_out, C);
}